// RegularizedChamferDistance_87514253623430
// MI455X (gfx1250) — compile-verified
//
#include <hip/hip_runtime.h>
#include <hip/hip_bf16.h>
#include <math.h>

typedef __attribute__((ext_vector_type(2))) float v2f;
typedef __attribute__((ext_vector_type(8))) float v8f;

#define NBATCH  2
#define NPTS    8192            // points per cloud (fixed by reference)
#define RADIUS2 0.01f
#define CSPLIT  4               // column-split factor (occupancy)
#define CHUNK_TILES ((NPTS / CSPLIT) / 16)   // 128 tiles per wave, compile-time

// ---------------- workspace layout (in 4-byte words) ----------------
#define TOTPTS   (NBATCH * NPTS)            // 16384
#define RMX_OFF  0                          // per-row min, dir X_hat->X (uint bits)
#define RMY_OFF  (RMX_OFF + TOTPTS)         // per-row min, dir X->X_hat
#define DENS_OFF (RMY_OFF + TOTPTS)         // densities (float)
#define NH_OFF   (DENS_OFF + TOTPTS)        // |X_hat|^2
#define NX_OFF   (NH_OFF + TOTPTS)          // |X|^2
#define PAH_OFF  (NX_OFF + TOTPTS)          // PackA(X_hat): 4 floats/pt
#define PBH_OFF  (PAH_OFF + 4*TOTPTS)       // PackB(X_hat)
#define PAX_OFF  (PBH_OFF + 4*TOTPTS)       // PackA(X)
#define PBX_OFF  (PAX_OFF + 4*TOTPTS)       // PackB(X)

// min(a,b) as a single v_med3 (no canonicalize traffic)
__device__ __forceinline__ float min_med3(float a, float b) {
    return __builtin_amdgcn_fmed3f(a, b, -INFINITY);
}
// xor-shuffle within groups of 32 lanes via ds_swizzle (imm must be literal)
template<int XORMASK>
__device__ __forceinline__ float swz_xor(float x) {
    return __int_as_float(
        __builtin_amdgcn_ds_swizzle(__float_as_int(x), (XORMASK << 10) | 0x1F));
}

__global__ void init_kernel(unsigned* rmx, unsigned* rmy, float* dens, int n) {
    int i = blockIdx.x * blockDim.x + threadIdx.x;
    if (i < n) {
        rmx[i] = 0x7F800000u;   // +inf
        rmy[i] = 0x7F800000u;
        dens[i] = 0.0f;
    }
}

// PackA[i] = (x, y, z, 1)           -> lane half0 loads (x,y), half1 loads (z,1)
// PackB[i] = (-2x, -2y, -2z, |p|^2) -> WMMA(A,B) = |b|^2 - 2 a.b
__global__ void pack_kernel(const float* __restrict__ X, float* __restrict__ norms,
                            float* __restrict__ packA, float* __restrict__ packB, int total) {
    int i = blockIdx.x * blockDim.x + threadIdx.x;
    if (i < total) {
        float x = X[3*(size_t)i], y = X[3*(size_t)i+1], z = X[3*(size_t)i+2];
        float n2 = x*x + y*y + z*z;
        norms[i] = n2;
        float* pa = packA + 4*(size_t)i;
        pa[0] = x;       pa[1] = y;       pa[2] = z;       pa[3] = 1.0f;
        float* pb = packB + 4*(size_t)i;
        pb[0] = -2.0f*x; pb[1] = -2.0f*y; pb[2] = -2.0f*z; pb[3] = n2;
    }
}

// One wave: 16-row tile of A x (NPTS/CSPLIT) columns of B.
// WMMA with C = |a_row|^2 yields d^2 directly; inner loop = load + wmma + 8 med3.
__global__ void __launch_bounds__(256)
chamfer_min_kernel(const float* __restrict__ packA, const float* __restrict__ normsA,
                   const float* __restrict__ packB, unsigned* __restrict__ rowmin)
{
    const int lane = threadIdx.x & 31;
    const int wave = (blockIdx.x * blockDim.x + threadIdx.x) >> 5;
    const int tilesPerBatch = NPTS >> 4;
    if (wave >= NBATCH * tilesPerBatch * CSPLIT) return;   // wave-uniform
    const int part = wave % CSPLIT;
    const int t    = wave / CSPLIT;
    const int b    = t / tilesPerBatch;
    const int row0 = (t % tilesPerBatch) * 16;
    const int half = lane >> 4;
    const int l15  = lane & 15;

    const v2f afrag = *(const v2f*)(packA + (((size_t)b*NPTS + row0 + l15) * 4 + half*2));

    v8f ca2;                                    // C operand: |a|^2 per row slot
    const float* na = normsA + (size_t)b*NPTS + row0 + half*8;
    #pragma unroll
    for (int v = 0; v < 8; ++v) ca2[v] = na[v];

    float mn[8];
    #pragma unroll
    for (int v = 0; v < 8; ++v) mn[v] = INFINITY;

    const float* bptr = packB +
        ((size_t)b*NPTS + (size_t)part*(NPTS/CSPLIT) + l15) * 4 + half*2;

    #pragma unroll 4
    for (int it = 0; it < CHUNK_TILES; ++it) {
        const v2f bfrag = *(const v2f*)bptr;
        bptr += 64;                             // 16 points * 4 floats
        v8f acc = __builtin_amdgcn_wmma_f32_16x16x4_f32(
                false, afrag, false, bfrag, (short)0, ca2, false, false);
        #pragma unroll
        for (int v = 0; v < 8; ++v)
            mn[v] = min_med3(mn[v], acc[v]);    // clamp deferred (monotone)
    }

    // per-row min across the 16 lanes of each half (xor masks stay in-half)
    #pragma unroll
    for (int v = 0; v < 8; ++v) {
        float m = mn[v];
        m = min_med3(m, swz_xor<1>(m));
        m = min_med3(m, swz_xor<2>(m));
        m = min_med3(m, swz_xor<4>(m));
        m = min_med3(m, swz_xor<8>(m));
        mn[v] = m;
    }
    if (l15 == 0) {                       // lane 0: rows v, lane 16: rows v+8
        unsigned* rm = rowmin + (size_t)b*NPTS + row0 + half*8;
        #pragma unroll
        for (int v = 0; v < 8; ++v) {
            float m = fmaxf(mn[v], 0.0f); // non-negative: uint order == float order
            atomicMin(&rm[v], __float_as_uint(m));
        }
    }
}

// Same tiling on (X_hat, X_hat); count d^2 <= RADIUS2 via ballot + scalar popcount.
__global__ void __launch_bounds__(256)
density_kernel(const float* __restrict__ packA, const float* __restrict__ normsA,
               const float* __restrict__ packB, float* __restrict__ dens)
{
    const int lane = threadIdx.x & 31;
    const int wave = (blockIdx.x * blockDim.x + threadIdx.x) >> 5;
    const int tilesPerBatch = NPTS >> 4;
    if (wave >= NBATCH * tilesPerBatch * CSPLIT) return;
    const int part = wave % CSPLIT;
    const int t    = wave / CSPLIT;
    const int b    = t / tilesPerBatch;
    const int row0 = (t % tilesPerBatch) * 16;
    const int half = lane >> 4;
    const int l15  = lane & 15;

    const v2f afrag = *(const v2f*)(packA + (((size_t)b*NPTS + row0 + l15) * 4 + half*2));

    v8f ca2;
    const float* na = normsA + (size_t)b*NPTS + row0 + half*8;
    #pragma unroll
    for (int v = 0; v < 8; ++v) ca2[v] = na[v];

    int cnt_lo[8], cnt_hi[8];                  // row v / row v+8 (wave-uniform, SGPRs)
    #pragma unroll
    for (int v = 0; v < 8; ++v) { cnt_lo[v] = 0; cnt_hi[v] = 0; }

    const float* bptr = packB +
        ((size_t)b*NPTS + (size_t)part*(NPTS/CSPLIT) + l15) * 4 + half*2;

    #pragma unroll 4
    for (int it = 0; it < CHUNK_TILES; ++it) {
        const v2f bfrag = *(const v2f*)bptr;
        bptr += 64;
        v8f acc = __builtin_amdgcn_wmma_f32_16x16x4_f32(
                false, afrag, false, bfrag, (short)0, ca2, false, false);
        #pragma unroll
        for (int v = 0; v < 8; ++v) {
            unsigned m = __builtin_amdgcn_ballot_w32(acc[v] <= RADIUS2);
            cnt_lo[v] += __builtin_popcount(m & 0xFFFFu);   // lanes 0-15 -> row v
            cnt_hi[v] += __builtin_popcount(m >> 16);       // lanes 16-31 -> row v+8
        }
    }

    if (lane == 0) {
        float* dp = dens + (size_t)b*NPTS + row0;
        #pragma unroll
        for (int v = 0; v < 8; ++v) {
            atomicAdd(&dp[v],     (float)cnt_lo[v]);
            atomicAdd(&dp[v + 8], (float)cnt_hi[v]);
        }
    }
}

// out = (sum rowminX + sum rowminY)/NPTS + 0.1 * std(dens, ddof=1)
__global__ void __launch_bounds__(256)
finalize_kernel(const float* __restrict__ rmx, const float* __restrict__ rmy,
                const float* __restrict__ dens, float* __restrict__ out)
{
    __shared__ float sc[256], ss[256], sq[256];
    float c = 0.0f, s = 0.0f, q = 0.0f;
    for (int i = threadIdx.x; i < TOTPTS; i += 256) {
        c += rmx[i] + rmy[i];
        float d = dens[i];
        s += d; q += d * d;
    }
    sc[threadIdx.x] = c; ss[threadIdx.x] = s; sq[threadIdx.x] = q;
    __syncthreads();
    for (int ofs = 128; ofs > 0; ofs >>= 1) {
        if (threadIdx.x < ofs) {
            sc[threadIdx.x] += sc[threadIdx.x + ofs];
            ss[threadIdx.x] += ss[threadIdx.x + ofs];
            sq[threadIdx.x] += sq[threadIdx.x + ofs];
        }
        __syncthreads();
    }
    if (threadIdx.x == 0) {
        const float n = (float)TOTPTS;
        float cham = sc[0] / (float)NPTS;
        float mean = ss[0] / n;
        float var  = (sq[0] - n * mean * mean) / (n - 1.0f);
        out[0] = cham + 0.1f * sqrtf(fmaxf(var, 0.0f));
    }
}

extern "C" void kernel_launch(void* const* d_in, const int* in_sizes, int n_in,
                              void* d_out, int out_size, void* d_ws, size_t ws_size,
                              hipStream_t stream) {
    const float* X_hat = (const float*)d_in[0];   // [2, 8192, 3]
    const float* X     = (const float*)d_in[1];   // [2, 8192, 3]
    float* out = (float*)d_out;
    float* ws  = (float*)d_ws;

    unsigned* rmx  = (unsigned*)(ws + RMX_OFF);
    unsigned* rmy  = (unsigned*)(ws + RMY_OFF);
    float* dens    = ws + DENS_OFF;
    float* normsH  = ws + NH_OFF;
    float* normsX  = ws + NX_OFF;
    float* packAH  = ws + PAH_OFF;
    float* packBH  = ws + PBH_OFF;
    float* packAX  = ws + PAX_OFF;
    float* packBX  = ws + PBX_OFF;

    const int nb = (TOTPTS + 255) / 256;          // 64 blocks
    init_kernel<<<nb, 256, 0, stream>>>(rmx, rmy, dens, TOTPTS);
    pack_kernel<<<nb, 256, 0, stream>>>(X_hat, normsH, packAH, packBH, TOTPTS);
    pack_kernel<<<nb, 256, 0, stream>>>(X,     normsX, packAX, packBX, TOTPTS);

    const int waves  = NBATCH * (NPTS >> 4) * CSPLIT;   // 4096 waves/pass
    const int blocks = (waves * 32 + 255) / 256;        // 512 blocks

    // cham_x: per X_hat point, min over X
    chamfer_min_kernel<<<blocks, 256, 0, stream>>>(packAH, normsH, packBX, rmx);
    // cham_y: per X point, min over X_hat
    chamfer_min_kernel<<<blocks, 256, 0, stream>>>(packAX, normsX, packBH, rmy);
    // densities over X_hat self-distances
    density_kernel<<<blocks, 256, 0, stream>>>(packAH, normsH, packBH, dens);

    finalize_kernel<<<1, 256, 0, stream>>>((const float*)rmx, (const float*)rmy,
                                           dens, out);
}